// Coulomb_SF_NB_47991964566178
// MI455X (gfx1250) — compile-verified
//
#include <hip/hip_runtime.h>

// Coulomb shifted-force neighbor-list energy for MI455X (gfx1250, wave32).
//
// Pass 1: one wave (32 lanes) per atom; lane L handles neighbor slots L, L+32.
//         Coalesced idx/mask loads, b96 gathers of coord[j], f32 math matching
//         the reference, 5-step shfl_xor wave reduce, per-block partial -> d_ws.
// Pass 2: single 1024-thread block reduces the 25000 block partials. The last
//         32-way cross-wave reduction is done with one V_WMMA_F32_16X16X4_F32
//         (A = wave sums with K1/K3 = 0, B = ones => D[m][*] = v[m]+v[m+16]).
//         Deterministic (fixed tree order, no float atomics); d_out overwritten.

typedef __attribute__((ext_vector_type(2))) float v2f;
typedef __attribute__((ext_vector_type(8))) float v8f;

#define WAVES_PER_BLOCK 8
#define BLOCK1 (WAVES_PER_BLOCK * 32)
#define KQQ 7.1998226f

__global__ __launch_bounds__(BLOCK1) void coul_sf_partial(
    const float* __restrict__ coord,
    const float* __restrict__ charges,
    const int* __restrict__ idxj,
    const unsigned char* __restrict__ padmask,   // XLA bool = 1 byte
    const float* __restrict__ rc_ptr,
    float* __restrict__ block_partial,
    int n_atoms, int m_nb)
{
    const int lane = threadIdx.x & 31;
    const int wid  = threadIdx.x >> 5;
    const int atom = blockIdx.x * WAVES_PER_BLOCK + wid;

    const float rc      = *rc_ptr;               // uniform -> scalar load
    const float inv_rc  = 1.0f / rc;
    const float inv_rc2 = inv_rc * inv_rc;

    float e  = 0.0f;
    float qi = 0.0f;
    if (atom < n_atoms) {
        qi = charges[atom];
        const float xi = coord[3 * atom + 0];
        const float yi = coord[3 * atom + 1];
        const float zi = coord[3 * atom + 2];
        const long long base = (long long)atom * m_nb;
        const float3* __restrict__ c3 = (const float3*)coord;   // 12B, 4B-aligned -> global_load_b96
        for (int mm = lane; mm < m_nb; mm += 32) {
            const int  j   = idxj[base + mm];
            const bool pad = padmask[base + mm] != 0;
            const float3 cj  = c3[j];
            const float  qj0 = charges[j];
            const float dx = cj.x - xi;
            const float dy = cj.y - yi;
            const float dz = cj.z - zi;
            float d2 = dx * dx + dy * dy + dz * dz;
            d2 = pad ? 1.0f : d2;                 // matches reference masking
            const float d  = sqrtf(d2);
            const float qj = (pad || (d > rc)) ? 0.0f : qj0;
            e += qj * (1.0f / d - inv_rc + (d - rc) * inv_rc2);
        }
    }

    // wave32 reduction: 5 xor-shuffle steps
    #pragma unroll
    for (int off = 16; off > 0; off >>= 1)
        e += __shfl_xor(e, off, 32);

    __shared__ float ssum[WAVES_PER_BLOCK];
    if (lane == 0) ssum[wid] = e * qi;            // q_i * sum_j term
    __syncthreads();
    if (threadIdx.x == 0) {
        float b = 0.0f;
        #pragma unroll
        for (int w = 0; w < WAVES_PER_BLOCK; ++w) b += ssum[w];
        block_partial[blockIdx.x] = b;
    }
}

__global__ __launch_bounds__(1024) void coul_sf_finish(
    const float* __restrict__ partial, int nblk, float* __restrict__ out)
{
    float a = 0.0f;
    for (int i = threadIdx.x; i < nblk; i += blockDim.x) a += partial[i];

    #pragma unroll
    for (int off = 16; off > 0; off >>= 1)
        a += __shfl_xor(a, off, 32);

    __shared__ float s[32];
    const int lane = threadIdx.x & 31;
    const int wid  = threadIdx.x >> 5;
    if (lane == 0) s[wid] = a;
    __syncthreads();

    if (wid == 0) {
        // 32 wave sums -> 1 via WMMA: A(16x4) holds v in K0 (lanes 0-15) and
        // K2 (lanes 16-31), zeros in K1/K3; B(4x16) = ones.
        // D[m][n] = v[m] + v[m+16]; per-lane sum of 8 D VGPRs covers M=0..7
        // (lanes<16) or M=8..15, so one shfl_xor(16) completes the total.
        v2f A; A.x = s[lane]; A.y = 0.0f;
        v2f B; B.x = 1.0f;    B.y = 1.0f;
        v8f C = {};
        v8f D = __builtin_amdgcn_wmma_f32_16x16x4_f32(
            /*neg_a=*/false, A, /*neg_b=*/false, B,
            /*c_mod=*/(short)0, C, /*reuse_a=*/false, /*reuse_b=*/false);
        float s8 = D[0] + D[1] + D[2] + D[3] + D[4] + D[5] + D[6] + D[7];
        s8 += __shfl_xor(s8, 16, 32);
        if (lane == 0) out[0] = KQQ * s8;
    }
}

extern "C" void kernel_launch(void* const* d_in, const int* in_sizes, int n_in,
                              void* d_out, int out_size, void* d_ws, size_t ws_size,
                              hipStream_t stream)
{
    (void)n_in; (void)out_size; (void)ws_size;
    const float*         coord   = (const float*)d_in[0];
    const float*         charges = (const float*)d_in[1];
    const int*           idxj    = (const int*)d_in[2];
    const unsigned char* mask    = (const unsigned char*)d_in[3];
    const float*         rc      = (const float*)d_in[4];

    const int n = in_sizes[1];            // atoms
    const int m = in_sizes[2] / n;        // neighbors per atom (64)
    const int nblk = (n + WAVES_PER_BLOCK - 1) / WAVES_PER_BLOCK;

    float* partials = (float*)d_ws;       // nblk floats (~100 KB)

    coul_sf_partial<<<nblk, BLOCK1, 0, stream>>>(coord, charges, idxj, mask, rc,
                                                 partials, n, m);
    coul_sf_finish<<<1, 1024, 0, stream>>>(partials, nblk, (float*)d_out);
}